// UCELossRegMap_15341623181346
// MI455X (gfx1250) — compile-verified
//
#include <hip/hip_runtime.h>

typedef __attribute__((ext_vector_type(2))) float v2f;
typedef __attribute__((ext_vector_type(4))) float v4f;
typedef __attribute__((ext_vector_type(8))) float v8f;

#define M_HEADS 4
#define HG 25
#define WG 25
#define NCAM 6
#define H0 56
#define W0 120
#define K_TOT (NCAM * H0 * W0)      /* 40320 */
#define CELLS (HG * WG)             /* 625   */
#define NB 2
#define TOTAL_CELLS (NB * CELLS)    /* 1250  */

// ---------------------------------------------------------------------------
// Kernel 1: per (b, cell), sum att0 over 4 heads via V_WMMA_F32_16X16X4_F32,
// argmax over K=40320. 1250 workgroups x 256 threads (8 wave32s).
//
// A = ones(16x4) -> D[m,n] = sum_h B[h,n] (identical rows) -> read d[0].
// Each lane loads float4 (global_load_b128, NT): half-wave = contiguous 256 B.
// WMMA j (j=0..3) uses component j of the two float4s as its B column, so
// column m of WMMA j carries k = k0 + 4m + j; per-lane index tracking keeps
// argmax exact. k ascends within each lane; ties resolved to smaller index
// in the cross-lane/wave reduction (matches jnp.argmax first-occurrence).
// ---------------------------------------------------------------------------
__global__ __launch_bounds__(256) void head_sum_argmax_kernel(
    const float* __restrict__ att0, int* __restrict__ out_idx) {
  const int cellG = blockIdx.x;            // 0..1249
  const int b = cellG / CELLS;
  const int c = cellG % CELLS;
  const int tid = threadIdx.x;
  const int wave = tid >> 5;               // 0..7
  const int lane = tid & 31;
  const int half = lane >> 4;              // 0: heads {0,1}, 1: heads {2,3}
  const int n = lane & 15;

  const size_t base0 =
      ((size_t)(b * M_HEADS + 2 * half) * CELLS + (size_t)c) * (size_t)K_TOT;
  const size_t base1 = base0 + (size_t)CELLS * (size_t)K_TOT;  // next head
  const float* __restrict__ p0 = att0 + base0;
  const float* __restrict__ p1 = att0 + base1;

  v2f a;
  a.x = 1.0f;
  a.y = 1.0f;

  float best = -3.402823466e+38f;
  int besti = 0;

  const int NCHUNK = K_TOT / 64;           // 630 chunks of 64 k-values
  for (int ch = wave; ch < NCHUNK; ch += 8) {
    const int k0 = ch * 64;
    // 16-byte aligned: base is a multiple of K_TOT floats, k0 % 64 == 0.
    v4f f0 = __builtin_nontemporal_load((const v4f*)(p0 + k0 + 4 * n));
    v4f f1 = __builtin_nontemporal_load((const v4f*)(p1 + k0 + 4 * n));
#pragma unroll
    for (int j = 0; j < 4; ++j) {
      v2f bm;
      bm.x = f0[j];
      bm.y = f1[j];
      v8f cz = {};
      v8f d = __builtin_amdgcn_wmma_f32_16x16x4_f32(
          /*neg_a=*/false, a, /*neg_b=*/false, bm,
          /*c_mod=*/(short)0, cz, /*reuse_a=*/false, /*reuse_b=*/false);
      float v = d[0];                      // 4-head sum for k = k0 + 4n + j
      int k = k0 + 4 * n + j;
      if (v > best) {                      // strict '>' keeps first occurrence
        best = v;
        besti = k;
      }
    }
  }

  // Intra-wave argmax reduction (wave32), tie-break on smaller index.
  for (int off = 16; off >= 1; off >>= 1) {
    float ov = __shfl_xor(best, off, 32);
    int oi = __shfl_xor(besti, off, 32);
    if (ov > best || (ov == best && oi < besti)) {
      best = ov;
      besti = oi;
    }
  }

  __shared__ float sv[8];
  __shared__ int si[8];
  if (lane == 0) {
    sv[wave] = best;
    si[wave] = besti;
  }
  __syncthreads();
  if (tid == 0) {
    float bv = sv[0];
    int bi = si[0];
    for (int w = 1; w < 8; ++w) {
      if (sv[w] > bv || (sv[w] == bv && si[w] < bi)) {
        bv = sv[w];
        bi = si[w];
      }
    }
    out_idx[cellG] = bi;
  }
}

// ---------------------------------------------------------------------------
// Kernel 2: per cell -> mask lookup, clipped logs, 8x8 sum of t, contribution.
// ---------------------------------------------------------------------------
__global__ void cell_loss_kernel(const int* __restrict__ idx_arr,
                                 const float* __restrict__ ood_cam,
                                 const float* __restrict__ y,
                                 float* __restrict__ contrib) {
  int t = blockIdx.x * blockDim.x + threadIdx.x;
  if (t >= TOTAL_CELLS) return;
  const int b = t / CELLS;
  const int c = t % CELLS;
  const int gi = c / WG;
  const int gj = c % WG;

  const int idx = idx_arr[t];
  const int mc = idx / (H0 * W0);
  const int rem = idx % (H0 * W0);
  const int my = rem / W0;
  const int mx = rem % W0;

  const float mask = ood_cam[((b * NCAM + mc) * H0 + my) * W0 + mx];
  const float logp = fmaxf(logf(mask), -100.0f);
  const float log1mp = fmaxf(log1pf(-mask), -100.0f);

  // t = y[b, 0, 8*gi : 8*gi+8, 8*gj : 8*gj+8]
  const float* __restrict__ yb = y + (size_t)(b * 4) * 200 * 200;
  float st = 0.0f;
  for (int r = 0; r < 8; ++r) {
    const float* __restrict__ row = yb + (size_t)(8 * gi + r) * 200 + 8 * gj;
    for (int cc = 0; cc < 8; ++cc) st += row[cc];
  }
  contrib[t] = st * logp + (64.0f - st) * log1mp;
}

// ---------------------------------------------------------------------------
// Kernel 3: deterministic single-block reduction -> reg = -sum / 160000.
// ---------------------------------------------------------------------------
__global__ __launch_bounds__(256) void final_reduce_kernel(
    const float* __restrict__ contrib, float* __restrict__ out) {
  __shared__ float s[256];
  const int tid = threadIdx.x;
  float acc = 0.0f;
  for (int t = tid; t < TOTAL_CELLS; t += 256) acc += contrib[t];
  s[tid] = acc;
  __syncthreads();
  for (int off = 128; off >= 1; off >>= 1) {
    if (tid < off) s[tid] += s[tid + off];
    __syncthreads();
  }
  if (tid == 0) out[0] = -s[0] / 160000.0f;
}

// ---------------------------------------------------------------------------
// Inputs (setup_inputs order): 0=alpha, 1=y, 2=ood, 3=ood_cam, 4=att0, 5=att1
// Only y, ood_cam, att0 are used by the reference result.
// ---------------------------------------------------------------------------
extern "C" void kernel_launch(void* const* d_in, const int* in_sizes, int n_in,
                              void* d_out, int out_size, void* d_ws,
                              size_t ws_size, hipStream_t stream) {
  const float* y = (const float*)d_in[1];
  const float* ood_cam = (const float*)d_in[3];
  const float* att0 = (const float*)d_in[4];

  int* ws_idx = (int*)d_ws;                                  // 1250 ints
  float* ws_contrib = (float*)((char*)d_ws + 8192);          // 1250 floats

  head_sum_argmax_kernel<<<TOTAL_CELLS, 256, 0, stream>>>(att0, ws_idx);
  cell_loss_kernel<<<(TOTAL_CELLS + 255) / 256, 256, 0, stream>>>(
      ws_idx, ood_cam, y, ws_contrib);
  final_reduce_kernel<<<1, 256, 0, stream>>>(ws_contrib, (float*)d_out);
}